// GINTopK4_88656714924040
// MI455X (gfx1250) — compile-verified
//
#include <hip/hip_runtime.h>
#include <math.h>

#define NN 50000
#define EE 600000
#define BB 256
#define HH 128
#define LDSP 132
#define BN_EPS 1e-5f

typedef __attribute__((ext_vector_type(2))) float v2f;
typedef __attribute__((ext_vector_type(8))) float v8f;

__device__ __forceinline__ int lower_bound_i(const int* __restrict__ a, int n, int v) {
    int lo = 0, hi = n;
    while (lo < hi) { int m = (lo + hi) >> 1; if (a[m] < v) lo = m + 1; else hi = m; }
    return lo;
}

// ---- init: copy x, set mask=1, zero out ----
__global__ void gin_init_kernel(const float* __restrict__ xin, float* __restrict__ xcur,
                                int* __restrict__ mask, float* __restrict__ out) {
    size_t idx = (size_t)blockIdx.x * blockDim.x + threadIdx.x;
    if (idx < (size_t)NN * HH) xcur[idx] = xin[idx];
    if (idx < NN) mask[idx] = 1;
    if (idx < (size_t)BB * 2 * HH) out[idx] = 0.0f;
}

// ---- per-layer zeroing of agg + BN accumulators ----
__global__ void gin_zero_kernel(float* __restrict__ agg, float* __restrict__ bnbuf) {
    size_t idx = (size_t)blockIdx.x * blockDim.x + threadIdx.x;
    if (idx < (size_t)NN * HH) agg[idx] = 0.0f;
    if (idx < 2 * HH + 1) bnbuf[idx] = 0.0f;
}

// ---- edge scatter-add: agg[dst] += x[src]; one wave per edge, float4 per lane ----
__global__ void gin_scatter_kernel(const float* __restrict__ x, const int* __restrict__ ei,
                                   float* __restrict__ agg) {
    size_t t = (size_t)blockIdx.x * blockDim.x + threadIdx.x;
    int e = (int)(t >> 5);
    int lane = (int)(t & 31);
    if (e >= EE) return;
    int s = ei[e];
    int d = ei[EE + e];
    const float4 v = ((const float4*)(x + (size_t)s * HH))[lane];
    float* ad = agg + (size_t)d * HH + lane * 4;
    atomicAdd(ad + 0, v.x);
    atomicAdd(ad + 1, v.y);
    atomicAdd(ad + 2, v.z);
    atomicAdd(ad + 3, v.w);
}

// ---- fused GIN MLP: h = relu((x+agg)@W1+b1)@W2+b2 via f32 WMMA ----
// block = 256 threads (8 waves) owns 16 rows; wave w owns output cols [16w,16w+16)
__global__ __launch_bounds__(256)
void gin_mlp_kernel(const float* __restrict__ x, const float* __restrict__ agg,
                    const float* __restrict__ W1, const float* __restrict__ b1,
                    const float* __restrict__ W2, const float* __restrict__ b2,
                    float* __restrict__ h) {
    __shared__ float h1s[16 * LDSP];
    const int r0   = blockIdx.x * 16;
    const int wave = threadIdx.x >> 5;
    const int lane = threadIdx.x & 31;
    const int hl   = lane >> 4;       // half of wave: 0 or 1
    const int l15  = lane & 15;
    const int c    = wave * 16 + l15; // output column of this lane
    const int koff = hl * 2;          // A/B K sub-offset per 16x16x4 f32 layout

    // ---- stage 1: h1 = relu((x+agg) @ W1 + b1), 16x16 tile ----
    v8f acc;
    {
        float bias = b1[c];
#pragma unroll
        for (int e = 0; e < 8; ++e) acc[e] = bias;
    }
    const size_t arow = (size_t)(r0 + l15) * HH;
    for (int k = 0; k < HH; k += 4) {
        const float2 xa = *(const float2*)(x + arow + k + koff);
        const float2 ga = *(const float2*)(agg + arow + k + koff);
        v2f a; a.x = xa.x + ga.x; a.y = xa.y + ga.y;
        v2f b;
        b.x = W1[(size_t)(k + koff) * HH + c];
        b.y = W1[(size_t)(k + koff + 1) * HH + c];
        acc = __builtin_amdgcn_wmma_f32_16x16x4_f32(false, a, false, b, (short)0, acc,
                                                    false, false);
    }
#pragma unroll
    for (int e = 0; e < 8; ++e) {
        int row = e + hl * 8; // D layout: vgpr e, lanes0-15 -> M=e, lanes16-31 -> M=e+8
        h1s[row * LDSP + c] = fmaxf(acc[e], 0.0f);
    }
    __syncthreads();

    // ---- stage 2: h = h1 @ W2 + b2 ----
    v8f acc2;
    {
        float bias = b2[c];
#pragma unroll
        for (int e = 0; e < 8; ++e) acc2[e] = bias;
    }
    for (int k = 0; k < HH; k += 4) {
        v2f a;
        a.x = h1s[l15 * LDSP + k + koff];
        a.y = h1s[l15 * LDSP + k + koff + 1];
        v2f b;
        b.x = W2[(size_t)(k + koff) * HH + c];
        b.y = W2[(size_t)(k + koff + 1) * HH + c];
        acc2 = __builtin_amdgcn_wmma_f32_16x16x4_f32(false, a, false, b, (short)0, acc2,
                                                     false, false);
    }
#pragma unroll
    for (int e = 0; e < 8; ++e) {
        int row = e + hl * 8;
        h[(size_t)(r0 + row) * HH + c] = acc2[e];
    }
}

// ---- BN stats over live nodes: 400 blocks x 125 rows, 128 threads (one col each) ----
__global__ void gin_bnstat_kernel(const float* __restrict__ h, const int* __restrict__ mask,
                                  float* __restrict__ bnbuf) {
    const int c = threadIdx.x;
    const int r0 = blockIdx.x * 125;
    float s = 0.0f, q = 0.0f;
    for (int r = 0; r < 125; ++r) {
        int node = r0 + r;
        float m = (float)mask[node];
        float v = h[(size_t)node * HH + c] * m;
        s += v;
        q += v * v;
    }
    atomicAdd(&bnbuf[c], s);
    atomicAdd(&bnbuf[HH + c], q);
    if (c == 0) {
        float cnt = 0.0f;
        for (int r = 0; r < 125; ++r) cnt += (float)mask[r0 + r];
        atomicAdd(&bnbuf[2 * HH], cnt);
    }
}

// ---- BN apply + relu + mask + pooling score (one block of 128 per node) ----
__global__ void gin_bnapply_kernel(float* __restrict__ h, const int* __restrict__ mask,
                                   const float* __restrict__ bnbuf,
                                   const float* __restrict__ gamma,
                                   const float* __restrict__ beta,
                                   const float* __restrict__ pw,
                                   float* __restrict__ score) {
    __shared__ float red[HH];
    const int node = blockIdx.x;
    const int c = threadIdx.x;
    float n = fmaxf(bnbuf[2 * HH], 1.0f);
    float mean = bnbuf[c] / n;
    float var = bnbuf[HH + c] / n - mean * mean;
    var = fmaxf(var, 0.0f);
    float v = h[(size_t)node * HH + c];
    v = (v - mean) * rsqrtf(var + BN_EPS) * gamma[c] + beta[c];
    v = fmaxf(v, 0.0f) * (float)mask[node];
    h[(size_t)node * HH + c] = v;

    float wc = pw[c];
    red[c] = v * wc;
    __syncthreads();
    for (int s = 64; s > 0; s >>= 1) {
        if (c < s) red[c] += red[c + s];
        __syncthreads();
    }
    float dot = red[0];
    __syncthreads();
    red[c] = wc * wc;
    __syncthreads();
    for (int s = 64; s > 0; s >>= 1) {
        if (c < s) red[c] += red[c + s];
        __syncthreads();
    }
    if (c == 0) score[node] = dot * rsqrtf(red[0]);
}

// ---- TopK per graph (block per graph). Rank matches stable lexsort(-s, batch). ----
__global__ void gin_topk_kernel(const float* __restrict__ h, const float* __restrict__ score,
                                const int* __restrict__ batch, const int* __restrict__ mask_in,
                                int* __restrict__ mask_out, float* __restrict__ xcur) {
    __shared__ int red[256];
    __shared__ int s_start, s_end;
    const int g = blockIdx.x;
    const int tid = threadIdx.x;
    if (tid == 0) {
        s_start = lower_bound_i(batch, NN, g);
        s_end = lower_bound_i(batch, NN, g + 1);
    }
    __syncthreads();
    const int start = s_start, end = s_end;

    int live = 0;
    for (int i = start + tid; i < end; i += 256) live += mask_in[i];
    red[tid] = live;
    __syncthreads();
    for (int s = 128; s > 0; s >>= 1) {
        if (tid < s) red[tid] += red[tid + s];
        __syncthreads();
    }
    const int total_live = red[0];
    const int k = (int)ceilf(0.8f * (float)total_live);

    for (int i = start + tid; i < end; i += 256) {
        int keep = 0;
        if (mask_in[i]) {
            float si = score[i];
            int rank = 0;
            for (int j = start; j < end; ++j) {
                if (mask_in[j]) {
                    float sj = score[j];
                    if (sj > si || (sj == si && j < i)) rank++;
                }
            }
            keep = (rank < k) ? 1 : 0;
        }
        mask_out[i] = keep;
        float t = keep ? tanhf(score[i]) : 0.0f;
        const float* hr = h + (size_t)i * HH;
        float* xr = xcur + (size_t)i * HH;
        for (int cc = 0; cc < HH; ++cc) xr[cc] = keep ? hr[cc] * t : 0.0f;
    }
}

// ---- global max + mean pool per graph; accumulate into out ----
__global__ void gin_pool_kernel(const float* __restrict__ x, const int* __restrict__ mask,
                                const int* __restrict__ batch, float* __restrict__ out) {
    __shared__ int s_start, s_end;
    const int g = blockIdx.x;
    const int c = threadIdx.x; // 128
    if (c == 0) {
        s_start = lower_bound_i(batch, NN, g);
        s_end = lower_bound_i(batch, NN, g + 1);
    }
    __syncthreads();
    float mx = -INFINITY, sum = 0.0f, cnt = 0.0f;
    for (int i = s_start; i < s_end; ++i) {
        if (mask[i]) {
            float v = x[(size_t)i * HH + c];
            mx = fmaxf(mx, v);
            sum += v;
            cnt += 1.0f;
        }
    }
    if (!(mx > -INFINITY)) mx = 0.0f;
    float mean = sum / fmaxf(cnt, 1.0f);
    out[(size_t)g * (2 * HH) + c] += mx;
    out[(size_t)g * (2 * HH) + HH + c] += mean;
}

extern "C" void kernel_launch(void* const* d_in, const int* in_sizes, int n_in,
                              void* d_out, int out_size, void* d_ws, size_t ws_size,
                              hipStream_t stream) {
    (void)in_sizes; (void)n_in; (void)out_size; (void)ws_size;
    const float* xin    = (const float*)d_in[0];
    const int*   ei     = (const int*)d_in[1];
    const int*   batch  = (const int*)d_in[2];
    const float* W1s    = (const float*)d_in[3];
    const float* b1s    = (const float*)d_in[4];
    const float* W2s    = (const float*)d_in[5];
    const float* b2s    = (const float*)d_in[6];
    const float* gammas = (const float*)d_in[7];
    const float* betas  = (const float*)d_in[8];
    const float* pws    = (const float*)d_in[9];
    float* out = (float*)d_out;

    const size_t NH = (size_t)NN * HH;
    float* fws   = (float*)d_ws;
    float* xcur  = fws;            // N*128
    float* agg   = xcur + NH;      // N*128
    float* h     = agg + NH;       // N*128
    float* score = h + NH;         // N
    float* bnbuf = score + NN;     // 257 (sums 128 | sumsq 128 | count)
    int*   maskA = (int*)(bnbuf + 260);
    int*   maskB = maskA + NN;

    dim3 b256(256), b128(128);
    const int gNH = (int)((NH + 255) / 256);

    gin_init_kernel<<<gNH, b256, 0, stream>>>(xin, xcur, maskA, out);

    for (int i = 0; i < 4; ++i) {
        int* m_in  = (i & 1) ? maskB : maskA;
        int* m_out = (i & 1) ? maskA : maskB;

        gin_zero_kernel<<<gNH, b256, 0, stream>>>(agg, bnbuf);
        gin_scatter_kernel<<<(EE * 32) / 256, b256, 0, stream>>>(xcur, ei, agg);
        gin_mlp_kernel<<<NN / 16, b256, 0, stream>>>(
            xcur, agg, W1s + (size_t)i * HH * HH, b1s + (size_t)i * HH,
            W2s + (size_t)i * HH * HH, b2s + (size_t)i * HH, h);
        gin_bnstat_kernel<<<400, b128, 0, stream>>>(h, m_in, bnbuf);
        gin_bnapply_kernel<<<NN, b128, 0, stream>>>(h, m_in, bnbuf, gammas + (size_t)i * HH,
                                                    betas + (size_t)i * HH,
                                                    pws + (size_t)i * HH, score);
        gin_topk_kernel<<<BB, b256, 0, stream>>>(h, score, batch, m_in, m_out, xcur);
        gin_pool_kernel<<<BB, b128, 0, stream>>>(xcur, m_out, batch, out);
    }
}